// MergedChunk12_25366076850540
// MI455X (gfx1250) — compile-verified
//
#include <hip/hip_runtime.h>

#define H    1024
#define NH   4
#define NKV  2
#define HD   256
#define DFF  2048
#define NL   15
#define NTOT 30
#define PLEW 256
#define WSL  1024
#define CTXL 4096
#define POS  100
#define EPSF 1e-6f

typedef float v2f __attribute__((ext_vector_type(2)));
typedef float v8f __attribute__((ext_vector_type(8)));

// ---------------------------------------------------------------------------
// One wave computes y[n0..n0+15] = sum_k x[k] * W[k*ldw + n0+j] with
// V_WMMA_F32_16X16X4_F32.  A-trick: all 16 rows of A carry the same x chunk,
// so D row 0 (== every row) holds the 16 dot products.
// 32-bit 16x4 A layout: lanes 0-15 K={0,1}, lanes 16-31 K={2,3}; B mirrors it.
// EXEC is full here (no divergent branches around the call sites).
// ---------------------------------------------------------------------------
__device__ __forceinline__ v8f wmma_gemv_acc(const float* __restrict__ x,
                                             const float* __restrict__ W,
                                             int K, int ldw, int n0, v8f acc) {
  const int lane = threadIdx.x & 31;
  const int hi   = lane >> 4;            // 0: lanes 0-15, 1: lanes 16-31
  const float* Wc = W + n0 + (lane & 15);
  for (int k0 = 0; k0 < K; k0 += 4) {
    v2f a, b;
    a.x = x[k0 + 2 * hi];
    a.y = x[k0 + 2 * hi + 1];
    b.x = Wc[(size_t)(k0 + 2 * hi) * ldw];
    b.y = Wc[(size_t)(k0 + 2 * hi + 1) * ldw];
    acc = __builtin_amdgcn_wmma_f32_16x16x4_f32(false, a, false, b,
                                                (short)0, acc, false, false);
  }
  return acc;
}

// blockDim == 256 (8 wave32).  Returns block-wide sum to all threads.
__device__ __forceinline__ float block_sum256(float v, float* red) {
  for (int off = 16; off; off >>= 1) v += __shfl_down(v, off);
  int t = threadIdx.x;
  if ((t & 31) == 0) red[t >> 5] = v;
  __syncthreads();
  float s = red[0] + red[1] + red[2] + red[3] + red[4] + red[5] + red[6] + red[7];
  __syncthreads();
  return s;
}

// Load h into LDS, RMS-normalize with weight lnw.
__device__ __forceinline__ void load_rmsnorm(const float* __restrict__ h,
                                             const float* __restrict__ lnw,
                                             float* x, float* red) {
  int t = threadIdx.x;
  float ss = 0.f;
  for (int i = t; i < H; i += 256) { float v = h[i]; x[i] = v; ss += v * v; }
  float tot = block_sum256(ss, red);
  float inv = rsqrtf(tot * (1.0f / H) + EPSF);
  for (int i = t; i < H; i += 256) x[i] = x[i] * inv * lnw[i];
  __syncthreads();
}

// ---------------------------------------------------------------------------
// Bulk KV-cache copy-through (outputs = inputs except row POS, patched later).
// ---------------------------------------------------------------------------
__global__ void copy4_kernel(const float4* __restrict__ src,
                             float4* __restrict__ dst, long n4) {
  long i = (long)blockIdx.x * blockDim.x + threadIdx.x;
  long stride = (long)gridDim.x * blockDim.x;
  for (; i < n4; i += stride) dst[i] = src[i];
}

__global__ void copy_vec_kernel(const float* __restrict__ src,
                                float* __restrict__ dst, int n) {
  for (int i = threadIdx.x; i < n; i += 256) dst[i] = src[i];
}

// ---------------------------------------------------------------------------
// PLE: proj = hs @ ple_proj_W * H^-0.5   (7680 outputs, WMMA GEMV)
// ---------------------------------------------------------------------------
__global__ void ple_proj_kernel(const float* __restrict__ hs,
                                const float* __restrict__ Wp,
                                float* __restrict__ proj) {
  __shared__ float x[H];
  int t = threadIdx.x;
  for (int i = t; i < H; i += 256) x[i] = hs[i];
  __syncthreads();
  int n0 = (blockIdx.x * 8 + (t >> 5)) * 16;
  v8f acc = {};
  acc = wmma_gemv_acc(x, Wp, H, NTOT * PLEW, n0, acc);
  if ((t & 31) < 16) proj[n0 + (t & 31)] = acc[0] * 0.03125f;  // 1/sqrt(1024)
}

// normed group RMS (the concat[pg,-pg] trick == mean(pg^2)), + raw, * 2^-0.5
__global__ void ple_norm_kernel(const float* __restrict__ proj,
                                const float* __restrict__ plr,
                                const float* __restrict__ pnw,
                                float* __restrict__ ple_ws,
                                float* __restrict__ ple_out) {
  __shared__ float red[8];
  int g = blockIdx.x, t = threadIdx.x;
  float v = proj[g * PLEW + t];
  float ms = block_sum256(v * v, red) * (1.0f / PLEW);
  float out = (v * rsqrtf(ms + EPSF) * pnw[t] + plr[g * PLEW + t]) * 0.70710678f;
  ple_ws[g * PLEW + t]  = out;
  ple_out[g * PLEW + t] = out;
}

// ---------------------------------------------------------------------------
// QKV projection: x = rms(h)*ln1;  [q|k|v] = x @ [Wq|Wk|Wv]  (2048 outputs)
// grid 16 x 256 : 128 waves x 16 cols
// ---------------------------------------------------------------------------
__global__ void qkv_kernel(const float* __restrict__ h,
                           const float* __restrict__ ln1_l,
                           const float* __restrict__ Wq_l,
                           const float* __restrict__ Wk_l,
                           const float* __restrict__ Wv_l,
                           float* __restrict__ qkv) {
  __shared__ float x[H];
  __shared__ float red[8];
  load_rmsnorm(h, ln1_l, x, red);
  int t = threadIdx.x;
  int out0 = (blockIdx.x * 8 + (t >> 5)) * 16;
  const float* W; int ldw, n0;
  if (out0 < 1024)      { W = Wq_l; ldw = NH * HD;  n0 = out0; }
  else if (out0 < 1536) { W = Wk_l; ldw = NKV * HD; n0 = out0 - 1024; }
  else                  { W = Wv_l; ldw = NKV * HD; n0 = out0 - 1536; }
  v8f acc = {};
  acc = wmma_gemv_acc(x, W, H, ldw, n0, acc);
  if ((t & 31) < 16) qkv[out0 + (t & 31)] = acc[0];
}

// ---------------------------------------------------------------------------
// Per-head q/k RMS + RoPE, write k/v into cache row POS, tap kv13/kv14.
// grid 8 x 256: blocks 0-3 q heads, 4-5 k heads, 6-7 v heads.
// ---------------------------------------------------------------------------
__global__ void rope_cache_kernel(const float* __restrict__ qkv,
                                  const float* __restrict__ qn_l,
                                  const float* __restrict__ kn_l,
                                  const float* __restrict__ cosb,
                                  const float* __restrict__ sinb,
                                  float* __restrict__ qrot,
                                  float* __restrict__ kcache,
                                  float* __restrict__ vcache,
                                  float* __restrict__ kv13_k,
                                  float* __restrict__ kv13_v,
                                  float* __restrict__ kv14_k,
                                  float* __restrict__ kv14_v) {
  __shared__ float n[HD];
  __shared__ float red[8];
  int b = blockIdx.x, t = threadIdx.x;
  if (b < NH) {                                   // q heads
    float v = qkv[b * HD + t];
    float ms = block_sum256(v * v, red) * (1.0f / HD);
    n[t] = v * rsqrtf(ms + EPSF) * qn_l[t];
    __syncthreads();
    float rot = (t < HD / 2) ? -n[t + HD / 2] : n[t - HD / 2];
    qrot[b * HD + t] = n[t] * cosb[t] + rot * sinb[t];
  } else if (b < NH + NKV) {                      // k heads
    int kh = b - NH;
    float v = qkv[NH * HD + kh * HD + t];
    float ms = block_sum256(v * v, red) * (1.0f / HD);
    n[t] = v * rsqrtf(ms + EPSF) * kn_l[t];
    __syncthreads();
    float rot = (t < HD / 2) ? -n[t + HD / 2] : n[t - HD / 2];
    float out = n[t] * cosb[t] + rot * sinb[t];
    kcache[(size_t)POS * (NKV * HD) + kh * HD + t] = out;
    if (kv13_k && kh == 0) kv13_k[t] = out;
    if (kv14_k) kv14_k[kh * HD + t] = out;
  } else {                                        // v heads (no norm/rope)
    int vh = b - NH - NKV;
    float v = qkv[(NH + NKV) * HD + vh * HD + t];
    vcache[(size_t)POS * (NKV * HD) + vh * HD + t] = v;
    if (kv13_v && vh == 0) kv13_v[t] = v;
    if (kv14_v) kv14_v[vh * HD + t] = v;
  }
}

// ---------------------------------------------------------------------------
// Attention.  Mask zeroes everything past POS exactly, so only rows 0..POS
// contribute.  grid NH x 256, head h reads kv-head h/2.
// ---------------------------------------------------------------------------
__global__ void attn_kernel(const float* __restrict__ q,
                            const float* __restrict__ Kc,
                            const float* __restrict__ Vc,
                            float* __restrict__ att) {
  __shared__ float qs[HD];
  __shared__ float p[128];
  __shared__ float red[8];
  int h = blockIdx.x, t = threadIdx.x;
  int kvh = h >> 1;
  qs[t] = q[h * HD + t];
  __syncthreads();
  float sc = -3.0e38f;
  if (t <= POS) {
    const float* krow = Kc + (size_t)t * (NKV * HD) + kvh * HD;
    float s = 0.f;
    #pragma unroll 8
    for (int d = 0; d < HD; ++d) s += qs[d] * krow[d];
    sc = s * 0.0625f;                 // HD^-0.5
  }
  // block max
  float m = sc;
  for (int off = 16; off; off >>= 1) m = fmaxf(m, __shfl_down(m, off));
  if ((t & 31) == 0) red[t >> 5] = m;
  __syncthreads();
  m = fmaxf(fmaxf(fmaxf(red[0], red[1]), fmaxf(red[2], red[3])),
            fmaxf(fmaxf(red[4], red[5]), fmaxf(red[6], red[7])));
  __syncthreads();
  float e = (t <= POS) ? __expf(sc - m) : 0.f;
  float ssum = block_sum256(e, red);
  if (t <= POS) p[t] = e / ssum;
  __syncthreads();
  float a = 0.f;
  const float* vb = Vc + kvh * HD + t;
  for (int r = 0; r <= POS; ++r) a += p[r] * vb[(size_t)r * (NKV * HD)];
  att[h * HD + t] = a;
}

// ---------------------------------------------------------------------------
// h += att @ Wo.  grid 8 x 256 -> 1024 outputs.
// ---------------------------------------------------------------------------
__global__ void oproj_kernel(const float* __restrict__ att,
                             const float* __restrict__ Wo_l,
                             float* __restrict__ hbuf) {
  __shared__ float x[H];
  int t = threadIdx.x;
  for (int i = t; i < H; i += 256) x[i] = att[i];
  __syncthreads();
  int n0 = (blockIdx.x * 8 + (t >> 5)) * 16;
  v8f acc = {};
  acc = wmma_gemv_acc(x, Wo_l, H, H, n0, acc);
  if ((t & 31) < 16) hbuf[n0 + (t & 31)] += acc[0];
}

// ---------------------------------------------------------------------------
// gate/up: x2 = rms(h)*ln2; gu = x2 @ [Wg|Wu].  grid 32 x 256 -> 4096 outputs.
// ---------------------------------------------------------------------------
__global__ void mlp_gu_kernel(const float* __restrict__ h,
                              const float* __restrict__ ln2_l,
                              const float* __restrict__ Wg_l,
                              const float* __restrict__ Wu_l,
                              float* __restrict__ gu) {
  __shared__ float x[H];
  __shared__ float red[8];
  load_rmsnorm(h, ln2_l, x, red);
  int t = threadIdx.x;
  int out0 = (blockIdx.x * 8 + (t >> 5)) * 16;
  const float* W = (out0 < DFF) ? Wg_l : Wu_l;
  int n0 = (out0 < DFF) ? out0 : out0 - DFF;
  v8f acc = {};
  acc = wmma_gemv_acc(x, W, H, DFF, n0, acc);
  if ((t & 31) < 16) gu[out0 + (t & 31)] = acc[0];
}

__global__ void mlp_act_kernel(const float* __restrict__ gu,
                               float* __restrict__ m) {
  int i = blockIdx.x * 256 + threadIdx.x;   // grid 8 x 256 == DFF
  float g = gu[i];
  float t = tanhf(0.79788456f * (g + 0.044715f * g * g * g));
  m[i] = 0.5f * g * (1.0f + t) * gu[DFF + i];
}

// ---------------------------------------------------------------------------
// h += m @ Wd + ple_l @ Wpl.  grid 8 x 256 -> 1024 outputs.
// ---------------------------------------------------------------------------
__global__ void mlp_down_kernel(const float* __restrict__ m,
                                const float* __restrict__ ple_l,
                                const float* __restrict__ Wd_l,
                                const float* __restrict__ Wpl_l,
                                float* __restrict__ hbuf) {
  __shared__ float xm[DFF];
  __shared__ float xp[PLEW];
  int t = threadIdx.x;
  for (int i = t; i < DFF; i += 256) xm[i] = m[i];
  for (int i = t; i < PLEW; i += 256) xp[i] = ple_l[i];
  __syncthreads();
  int n0 = (blockIdx.x * 8 + (t >> 5)) * 16;
  v8f acc = {};
  acc = wmma_gemv_acc(xm, Wd_l, DFF, H, n0, acc);
  acc = wmma_gemv_acc(xp, Wpl_l, PLEW, H, n0, acc);
  if ((t & 31) < 16) hbuf[n0 + (t & 31)] += acc[0];
}

// ---------------------------------------------------------------------------
extern "C" void kernel_launch(void* const* d_in, const int* in_sizes, int n_in,
                              void* d_out, int out_size, void* d_ws, size_t ws_size,
                              hipStream_t stream) {
  (void)in_sizes; (void)n_in; (void)out_size; (void)ws_size;
  const float* hs    = (const float*)d_in[0];
  const float* cos_s = (const float*)d_in[5];
  const float* sin_s = (const float*)d_in[6];
  const float* Ks_in = (const float*)d_in[9];
  const float* Vs_in = (const float*)d_in[10];
  const float* Kf_in = (const float*)d_in[11];
  const float* Vf_in = (const float*)d_in[12];
  const float* Wq  = (const float*)d_in[13];
  const float* Wk  = (const float*)d_in[14];
  const float* Wv  = (const float*)d_in[15];
  const float* Wo  = (const float*)d_in[16];
  const float* Wg  = (const float*)d_in[17];
  const float* Wu  = (const float*)d_in[18];
  const float* Wd  = (const float*)d_in[19];
  const float* ln1 = (const float*)d_in[20];
  const float* ln2 = (const float*)d_in[21];
  const float* qn  = (const float*)d_in[22];
  const float* kn  = (const float*)d_in[23];
  const float* Wpl = (const float*)d_in[24];
  const float* Wpp = (const float*)d_in[25];
  const float* pnw = (const float*)d_in[26];
  const float* plr = (const float*)d_in[4];

  const size_t SLID = 12ul * WSL * (NKV * HD);   // 6,291,456
  const size_t FULL = 3ul  * CTXL * (NKV * HD);  // 6,291,456
  float* out   = (float*)d_out;
  float* o_h   = out;
  float* o_Ks  = out + 1024;
  float* o_Vs  = o_Ks + SLID;
  float* o_Kf  = o_Vs + SLID;
  float* o_Vf  = o_Kf + FULL;
  float* o_k13 = o_Vf + FULL;
  float* o_v13 = o_k13 + 256;
  float* o_k14 = o_v13 + 256;
  float* o_v14 = o_k14 + 512;
  float* o_ple = o_v14 + 512;

  float* ws     = (float*)d_ws;
  float* w_h    = ws;          // 1024
  float* w_qkv  = ws + 1024;   // 2048
  float* w_qrot = ws + 3072;   // 1024
  float* w_att  = ws + 4096;   // 1024
  float* w_gu   = ws + 5120;   // 4096
  float* w_m    = ws + 9216;   // 2048
  float* w_proj = ws + 11264;  // 7680
  float* w_ple  = ws + 18944;  // 7680

  // KV-cache pass-through (row POS patched per-layer afterwards, in order)
  copy4_kernel<<<2048, 256, 0, stream>>>((const float4*)Ks_in, (float4*)o_Ks, (long)(SLID / 4));
  copy4_kernel<<<2048, 256, 0, stream>>>((const float4*)Vs_in, (float4*)o_Vs, (long)(SLID / 4));
  copy4_kernel<<<2048, 256, 0, stream>>>((const float4*)Kf_in, (float4*)o_Kf, (long)(FULL / 4));
  copy4_kernel<<<2048, 256, 0, stream>>>((const float4*)Vf_in, (float4*)o_Vf, (long)(FULL / 4));

  copy_vec_kernel<<<1, 256, 0, stream>>>(hs, w_h, H);
  ple_proj_kernel<<<60, 256, 0, stream>>>(hs, Wpp, w_proj);
  ple_norm_kernel<<<NTOT, 256, 0, stream>>>(w_proj, plr, pnw, w_ple, o_ple);

  int si = 0, fi = 0;
  for (int l = 0; l < NL; ++l) {
    bool full = (l == 4 || l == 9 || l == 14);
    float *kc, *vc;
    if (full) { kc = o_Kf + (size_t)fi * CTXL * (NKV * HD);
                vc = o_Vf + (size_t)fi * CTXL * (NKV * HD); ++fi; }
    else      { kc = o_Ks + (size_t)si * WSL  * (NKV * HD);
                vc = o_Vs + (size_t)si * WSL  * (NKV * HD); ++si; }

    qkv_kernel<<<16, 256, 0, stream>>>(w_h, ln1 + (size_t)l * H,
        Wq + (size_t)l * H * (NH * HD), Wk + (size_t)l * H * (NKV * HD),
        Wv + (size_t)l * H * (NKV * HD), w_qkv);
    rope_cache_kernel<<<8, 256, 0, stream>>>(w_qkv, qn + (size_t)l * HD,
        kn + (size_t)l * HD, cos_s, sin_s, w_qrot, kc, vc,
        (l == 13) ? o_k13 : nullptr, (l == 13) ? o_v13 : nullptr,
        (l == 14) ? o_k14 : nullptr, (l == 14) ? o_v14 : nullptr);
    attn_kernel<<<NH, 256, 0, stream>>>(w_qrot, kc, vc, w_att);
    oproj_kernel<<<8, 256, 0, stream>>>(w_att, Wo + (size_t)l * (NH * HD) * H, w_h);
    mlp_gu_kernel<<<32, 256, 0, stream>>>(w_h, ln2 + (size_t)l * H,
        Wg + (size_t)l * H * DFF, Wu + (size_t)l * H * DFF, w_gu);
    mlp_act_kernel<<<8, 256, 0, stream>>>(w_gu, w_m);
    mlp_down_kernel<<<8, 256, 0, stream>>>(w_m, w_ple + (size_t)l * PLEW,
        Wd + (size_t)l * DFF * H, Wpl + (size_t)l * PLEW * H, w_h);
  }
  copy_vec_kernel<<<1, 256, 0, stream>>>(w_h, o_h, H);
}